// Net_30554397343961
// MI455X (gfx1250) — compile-verified
//
#include <hip/hip_runtime.h>
#include <math.h>

// ---------------------------------------------------------------------------
// Problem constants (match reference): B=1024, K=256, S=50000, E=20000
// ---------------------------------------------------------------------------
#define BSZ   1024
#define KDIM  256
#define INV_SOFT 5.0f      // 1/0.2
#define LPAD  260          // LDS row pitch (floats): multiple of 4 for b128,
                           // 260 % 64 == 4 -> rows staggered by 4 banks

typedef float v2f __attribute__((ext_vector_type(2)));
typedef float v8f __attribute__((ext_vector_type(8)));

// ---------------------------------------------------------------------------
// Workspace layout (floats)
// ---------------------------------------------------------------------------
#define OFF_STU_RAW   0                         // [B,K]
#define OFF_A_NORM    (OFF_STU_RAW  + BSZ*KDIM) // [B,K]
#define OFF_B_NORM    (OFF_A_NORM   + BSZ*KDIM) // [B,K]
#define OFF_EXER_G    (OFF_B_NORM   + BSZ*KDIM) // [B,K]
#define OFF_STU_PART  (OFF_EXER_G   + BSZ*KDIM) // [B,K]
#define OFF_EXER_PART (OFF_STU_PART + BSZ*KDIM) // [B,K]
#define OFF_KN1       (OFF_EXER_PART+ BSZ*KDIM) // [K,K]
#define OFF_KN2       (OFF_KN1      + KDIM*KDIM)// [K,K]
#define OFF_NROW      (OFF_KN2      + KDIM*KDIM)// [B]
#define OFF_DIAG      (OFF_NROW     + BSZ)      // [B]
#define OFF_PRS       (OFF_DIAG     + BSZ)      // [64,B]  per-tile row sums
#define OFF_POUT      (OFF_PRS      + 64*BSZ)   // [32,B]  per-i-tile out sums

// ---------------------------------------------------------------------------
// One 16x16 f32 tile of C = A(row-major,lda) @ Brm(row-major,ldb)^T, K=256,
// accumulated with V_WMMA_F32_16X16X4_F32.  One wave per tile.
// A 16x4 layout: lane = M + 16*(K>=2), vgpr = K%2  -> lane loads float2 at k
// B 4x16 layout: lane = N + 16*(K>=2), vgpr = K%2  -> lane loads Brm[n][k..k+1]
// C/D: vgpr r holds row M=r (lanes 0-15) and M=r+8 (lanes 16-31), N=lane%16.
// ---------------------------------------------------------------------------
__device__ __forceinline__ v8f wmma_tile_abT(const float* __restrict__ A, int lda,
                                             const float* __restrict__ Brm, int ldb,
                                             int tm, int tn) {
  const int lane = threadIdx.x & 31;
  const int l = lane & 15;
  const int h = lane >> 4;
  const float* ap = A   + (tm * 16 + l) * lda + 2 * h;
  const float* bp = Brm + (tn * 16 + l) * ldb + 2 * h;
  v8f acc = {};
#pragma unroll 8
  for (int k0 = 0; k0 < KDIM; k0 += 4) {
    v2f av = *(const v2f*)(ap + k0);
    v2f bv = *(const v2f*)(bp + k0);
    acc = __builtin_amdgcn_wmma_f32_16x16x4_f32(false, av, false, bv,
                                                (short)0, acc, false, false);
  }
  return acc;
}

// ---------------------------------------------------------------------------
// Kernel 1: gather rows, L2-normalize, gather exercise rows, compute n_row.
// One block per batch element b, 256 threads = one k element each.
// ---------------------------------------------------------------------------
__global__ void __launch_bounds__(256)
gather_norm_kernel(const int* __restrict__ stu_id, const int* __restrict__ exer_id,
                   const float* __restrict__ stu_emb, const float* __restrict__ stu_emb_per,
                   const float* __restrict__ exer_emb,
                   float* __restrict__ stu_raw, float* __restrict__ a_norm,
                   float* __restrict__ b_norm,  float* __restrict__ exer_g,
                   float* __restrict__ nrow) {
  __shared__ float red[256];
  const int b = blockIdx.x;
  const int t = threadIdx.x;
  const int sid = stu_id[b];
  const int eid = exer_id[b];

  const float x = stu_emb[sid * KDIM + t];
  const float y = stu_emb_per[sid * KDIM + t];
  stu_raw[b * KDIM + t] = x;
  exer_g[b * KDIM + t]  = exer_emb[eid * KDIM + t];

  red[t] = x * x;
  __syncthreads();
  for (int s = 128; s > 0; s >>= 1) { if (t < s) red[t] += red[t + s]; __syncthreads(); }
  const float nx = sqrtf(red[0]);
  __syncthreads();

  red[t] = y * y;
  __syncthreads();
  for (int s = 128; s > 0; s >>= 1) { if (t < s) red[t] += red[t + s]; __syncthreads(); }
  const float ny = sqrtf(red[0]);

  const float dx = fmaxf(nx, 1e-12f);
  const float dy = fmaxf(ny, 1e-12f);
  a_norm[b * KDIM + t] = x / dx;
  b_norm[b * KDIM + t] = y / dy;
  if (t == 0) nrow[b] = nx / dx;   // norm of the already-normalized a row
}

// ---------------------------------------------------------------------------
// Kernel 2: sim = a @ b^T via WMMA, fused scale 1/(n_m*n_n*SOFT),
// deterministic per-tile row-sum partials + diagonal extraction.
// 64x64 tiles of 16x16; one wave per tile; 8 waves/block -> grid 512.
// ---------------------------------------------------------------------------
__global__ void __launch_bounds__(256)
sim_kernel(const float* __restrict__ a, const float* __restrict__ b,
           const float* __restrict__ nrow,
           float* __restrict__ part_rowsum,   // [64][1024] indexed [tn][m]
           float* __restrict__ diag_s) {
  const int wave = (blockIdx.x * blockDim.x + threadIdx.x) >> 5;
  const int tm = wave >> 6;
  const int tn = wave & 63;
  v8f acc = wmma_tile_abT(a, KDIM, b, KDIM, tm, tn);

  const int lane = threadIdx.x & 31;
  const int l = lane & 15;
  const int h = lane >> 4;
  const float nn = nrow[tn * 16 + l];

#pragma unroll
  for (int r = 0; r < 8; ++r) {
    const int m = tm * 16 + 8 * h + r;
    // n ~= 1 by construction; fast v_rcp_f32 instead of IEEE divide sequence
    const float val = acc[r] * __builtin_amdgcn_rcpf(nrow[m] * nn) * INV_SOFT;
    if (tm == tn && l == 8 * h + r) diag_s[m] = val;   // diagonal element
    // sum over the 16 lanes that hold row m (same half h)
    float s = val;
    s += __shfl_xor(s, 1, 32);
    s += __shfl_xor(s, 2, 32);
    s += __shfl_xor(s, 4, 32);
    s += __shfl_xor(s, 8, 32);
    if (l == 0) part_rowsum[tn * BSZ + m] = s;
  }
}

// ---------------------------------------------------------------------------
// Kernel 3: generic C[M,N] = A @ Brm^T (all row-major, K=256) via WMMA.
// One wave per 16x16 tile; tilesN = N/16.
// ---------------------------------------------------------------------------
__global__ void __launch_bounds__(256)
gemm_abT_kernel(const float* __restrict__ A, int lda,
                const float* __restrict__ Brm, int ldb,
                float* __restrict__ C, int ldc, int tilesN) {
  const int wave = (blockIdx.x * blockDim.x + threadIdx.x) >> 5;
  const int tm = wave / tilesN;
  const int tn = wave % tilesN;
  v8f acc = wmma_tile_abT(A, lda, Brm, ldb, tm, tn);

  const int lane = threadIdx.x & 31;
  const int l = lane & 15;
  const int h = lane >> 4;
#pragma unroll
  for (int r = 0; r < 8; ++r)
    C[(tm * 16 + 8 * h + r) * ldc + tn * 16 + l] = acc[r];
}

// ---------------------------------------------------------------------------
// Kernel 4: the transcendental-bound core.
//   s[b,i]   = sum_j (sig(stu[b,j]+kn1[i,j]) - sig(exer[b,j]+kn2[i,j]))*W3[j]
//   contrib  = kn_r[b,i] * sig(s[b,i]+b3)
// Tile: 16 b x 8 i per 128-thread block; all operands staged in LDS with
// pitch 260 floats so every stream is read with ds_load_b128, conflict-free.
// Partial sums over the i-tile go deterministically to part_out[it][b].
// ---------------------------------------------------------------------------
__device__ __forceinline__ float sigf(float x) {
  return 1.f / (1.f + __expf(-x));
}

__global__ void __launch_bounds__(128)
fuse_kernel(const float* __restrict__ stu_part, const float* __restrict__ exer_part,
            const float* __restrict__ kn1, const float* __restrict__ kn2,
            const float* __restrict__ W3, const float* __restrict__ b3,
            const float* __restrict__ kn_r,
            float* __restrict__ part_out) {   // [32][1024] indexed [it][b]
  __shared__ float s_stu [16 * LPAD];
  __shared__ float s_exer[16 * LPAD];
  __shared__ float s_kn1 [ 8 * LPAD];
  __shared__ float s_kn2 [ 8 * LPAD];
  __shared__ float s_w3  [256];
  __shared__ float s_red [128];

  const int bt = blockIdx.x >> 5;   // 64 b-tiles
  const int it = blockIdx.x & 31;   // 32 i-tiles
  const int t  = threadIdx.x;

  // stage with b128 loads/stores: 64 float4 per row
  for (int idx = t; idx < 16 * 64; idx += 128) {
    const int r = idx >> 6, c4 = idx & 63;
    ((float4*)&s_stu [r * LPAD])[c4] = ((const float4*)&stu_part [(bt * 16 + r) * KDIM])[c4];
    ((float4*)&s_exer[r * LPAD])[c4] = ((const float4*)&exer_part[(bt * 16 + r) * KDIM])[c4];
  }
  for (int idx = t; idx < 8 * 64; idx += 128) {
    const int r = idx >> 6, c4 = idx & 63;
    ((float4*)&s_kn1[r * LPAD])[c4] = ((const float4*)&kn1[(it * 8 + r) * KDIM])[c4];
    ((float4*)&s_kn2[r * LPAD])[c4] = ((const float4*)&kn2[(it * 8 + r) * KDIM])[c4];
  }
  for (int idx = t; idx < 64; idx += 128)
    ((float4*)s_w3)[idx] = ((const float4*)W3)[idx];
  __syncthreads();

  const int bb = t & 15;   // 16 b per block
  const int ii = t >> 4;   // 8 i per block
  const float4* ps = (const float4*)&s_stu [bb * LPAD];
  const float4* pe = (const float4*)&s_exer[bb * LPAD];
  const float4* p1 = (const float4*)&s_kn1 [ii * LPAD];
  const float4* p2 = (const float4*)&s_kn2 [ii * LPAD];
  const float4* pw = (const float4*)s_w3;

  float acc = 0.f;
#pragma unroll 4
  for (int j4 = 0; j4 < KDIM / 4; ++j4) {
    const float4 a = ps[j4], e = pe[j4], k1 = p1[j4], k2 = p2[j4], w = pw[j4];
    acc += (sigf(a.x + k1.x) - sigf(e.x + k2.x)) * w.x;
    acc += (sigf(a.y + k1.y) - sigf(e.y + k2.y)) * w.y;
    acc += (sigf(a.z + k1.z) - sigf(e.z + k2.z)) * w.z;
    acc += (sigf(a.w + k1.w) - sigf(e.w + k2.w)) * w.w;
  }
  const float o = sigf(acc + b3[0]);
  const int bg = bt * 16 + bb;
  const int ig = it * 8 + ii;
  s_red[t] = kn_r[bg * KDIM + ig] * o;
  __syncthreads();
  // reduce over ii (stride-16 groups): strides 64, 32, 16
  for (int s = 64; s >= 16; s >>= 1) {
    if (t < s) s_red[t] += s_red[t + s];
    __syncthreads();
  }
  if (t < 16) part_out[it * BSZ + bt * 16 + t] = s_red[t];
}

// ---------------------------------------------------------------------------
// Kernel 5: output[b] = (sum_it part_out[it][b]) / (sum_i kn_r[b,i])
// ---------------------------------------------------------------------------
__global__ void __launch_bounds__(256)
finalize_out_kernel(const float* __restrict__ part_out,
                    const float* __restrict__ kn_r,
                    float* __restrict__ out) {
  const int b = blockIdx.x * 256 + threadIdx.x;
  float s = 0.f;
#pragma unroll
  for (int it = 0; it < 32; ++it) s += part_out[it * BSZ + b];
  float cnt = 0.f;
  const float4* kr = (const float4*)&kn_r[b * KDIM];
#pragma unroll 8
  for (int i4 = 0; i4 < KDIM / 4; ++i4) {
    const float4 v = kr[i4];
    cnt += v.x + v.y + v.z + v.w;
  }
  out[b] = s / cnt;
}

// ---------------------------------------------------------------------------
// Kernel 6: ssl_loss = sum_b -log(max(exp(diag[b]) / (rowsum[b]+1e-8), 1e-8))
// ---------------------------------------------------------------------------
__global__ void __launch_bounds__(256)
ssl_kernel(const float* __restrict__ diag_s,
           const float* __restrict__ part_rowsum,
           float* __restrict__ out_loss) {
  __shared__ float red[256];
  const int t = threadIdx.x;
  float loc = 0.f;
  for (int b = t; b < BSZ; b += 256) {
    float rs = 0.f;
#pragma unroll 8
    for (int tn = 0; tn < 64; ++tn) rs += part_rowsum[tn * BSZ + b];
    const float divided = __expf(diag_s[b]) / (rs + 1e-8f);
    loc += -__logf(fmaxf(divided, 1e-8f));
  }
  red[t] = loc;
  __syncthreads();
  for (int s = 128; s > 0; s >>= 1) { if (t < s) red[t] += red[t + s]; __syncthreads(); }
  if (t == 0) out_loss[0] = red[0];
}

// ---------------------------------------------------------------------------
// Launch
// ---------------------------------------------------------------------------
extern "C" void kernel_launch(void* const* d_in, const int* in_sizes, int n_in,
                              void* d_out, int out_size, void* d_ws, size_t ws_size,
                              hipStream_t stream) {
  const int*   stu_id   = (const int*)  d_in[0];
  const int*   exer_id  = (const int*)  d_in[1];
  const float* kn_r     = (const float*)d_in[2];
  const float* stu_emb  = (const float*)d_in[3];
  const float* stu_embp = (const float*)d_in[4];
  const float* exer_emb = (const float*)d_in[5];
  const float* kn_emb2  = (const float*)d_in[6];
  const float* W1       = (const float*)d_in[7];   // [K, 2K]
  const float* W2       = (const float*)d_in[8];   // [K, 2K]
  const float* W3       = (const float*)d_in[9];   // [1, K]
  const float* b3       = (const float*)d_in[10];  // [1]
  float* out = (float*)d_out;                      // [1024] output + [1] ssl_loss
  float* ws  = (float*)d_ws;

  float* stu_raw   = ws + OFF_STU_RAW;
  float* a_norm    = ws + OFF_A_NORM;
  float* b_norm    = ws + OFF_B_NORM;
  float* exer_g    = ws + OFF_EXER_G;
  float* stu_part  = ws + OFF_STU_PART;
  float* exer_part = ws + OFF_EXER_PART;
  float* kn1       = ws + OFF_KN1;
  float* kn2       = ws + OFF_KN2;
  float* nrow      = ws + OFF_NROW;
  float* diag_s    = ws + OFF_DIAG;
  float* prs       = ws + OFF_PRS;
  float* pout      = ws + OFF_POUT;

  // 1) gather + normalize
  gather_norm_kernel<<<BSZ, 256, 0, stream>>>(stu_id, exer_id, stu_emb, stu_embp,
                                              exer_emb, stu_raw, a_norm, b_norm,
                                              exer_g, nrow);
  // 2) sim GEMM + fused SSL partials (64x64 tiles, 8 waves/block)
  sim_kernel<<<512, 256, 0, stream>>>(a_norm, b_norm, nrow, prs, diag_s);
  // 3) four A@W^T GEMMs (f32 WMMA)
  gemm_abT_kernel<<<128, 256, 0, stream>>>(stu_raw, KDIM, W1,        2 * KDIM, stu_part,  KDIM, KDIM / 16);
  gemm_abT_kernel<<<128, 256, 0, stream>>>(exer_g,  KDIM, W2,        2 * KDIM, exer_part, KDIM, KDIM / 16);
  gemm_abT_kernel<<< 32, 256, 0, stream>>>(kn_emb2, KDIM, W1 + KDIM, 2 * KDIM, kn1,       KDIM, KDIM / 16);
  gemm_abT_kernel<<< 32, 256, 0, stream>>>(kn_emb2, KDIM, W2 + KDIM, 2 * KDIM, kn2,       KDIM, KDIM / 16);
  // 4) sigmoid-difference reduction over [B,K,K]
  fuse_kernel<<<(BSZ / 16) * (KDIM / 8), 128, 0, stream>>>(stu_part, exer_part, kn1, kn2,
                                                           W3, b3, kn_r, pout);
  // 5) outputs
  finalize_out_kernel<<<BSZ / 256, 256, 0, stream>>>(pout, kn_r, out);
  ssl_kernel<<<1, 256, 0, stream>>>(diag_s, prs, out + BSZ);
}